// GlyphBag_36395552867098
// MI455X (gfx1250) — compile-verified
//
#include <hip/hip_runtime.h>
#include <hip/hip_bf16.h>

typedef __attribute__((ext_vector_type(2))) float v2f;
typedef __attribute__((ext_vector_type(8))) float v8f;

#define GLYPH_DIM 4096
#define PAD_IDX   4096
#define EMB_DIM   20
#define HIDDEN    32
#define MAX_LEN   64
#define HW_CELLS  (21 * 79)

// -------------------------------------------------------------------------
// Kernel 1: per-sample presence bitmap -> sorted-unique bag -> embeddings.
// One 256-thread workgroup per sample; 109 MB of glyph reads dominate the
// whole problem (~5 us at 23.3 TB/s), compaction is done entirely in LDS.
// -------------------------------------------------------------------------
__global__ __launch_bounds__(256) void glyph_bag_kernel(
    const int* __restrict__ chars, const int* __restrict__ colors,
    const float* __restrict__ embT,   // [4097][20], row 4096 == 0
    float* __restrict__ outEmb,       // [B][64][20]
    float* __restrict__ outBag,       // [B][64] (float-encoded ids)
    int*   __restrict__ lenv)         // [B] scratch lengths
{
  __shared__ unsigned bm[128];
  __shared__ int scan[2][128];        // ping-pong inclusive scan of popcounts
  __shared__ int bagL[MAX_LEN];

  const int b   = blockIdx.x;
  const int tid = threadIdx.x;

  if (tid < 128) bm[tid] = 0u;
  __syncthreads();

  const int* cr = chars  + (size_t)b * HW_CELLS;
  const int* co = colors + (size_t)b * HW_CELLS;
  for (int i = tid; i < HW_CELLS; i += 256) {
    int g = (cr[i] << 4) + co[i];     // 0..4095
    atomicOr(&bm[g >> 5], 1u << (g & 31));
  }
  __syncthreads();

  // Hillis-Steele inclusive scan of per-word popcounts (7 steps over 128)
  if (tid < 128) scan[0][tid] = __popc(bm[tid]);
  __syncthreads();
  int src = 0;
#pragma unroll
  for (int d = 1; d < 128; d <<= 1) {
    if (tid < 128) {
      int v = scan[src][tid];
      if (tid >= d) v += scan[src][tid - d];
      scan[src ^ 1][tid] = v;
    }
    src ^= 1;
    __syncthreads();
  }
  if (tid < MAX_LEN) bagL[tid] = PAD_IDX;
  __syncthreads();

  if (tid < 128) {                    // emit set bits in ascending order
    unsigned w = bm[tid];
    int r = scan[src][tid] - __popc(w);          // exclusive prefix
    while (w && r < MAX_LEN) {
      int bit = __builtin_ctz(w);
      bagL[r] = (tid << 5) + bit;
      w &= w - 1u;
      ++r;
    }
  }
  __syncthreads();

  if (tid == 0) {
    int total = scan[src][127];
    lenv[b] = total < MAX_LEN ? total : MAX_LEN;
  }
  if (tid < MAX_LEN) outBag[(size_t)b * MAX_LEN + tid] = (float)bagL[tid];

  for (int i = tid; i < MAX_LEN * EMB_DIM; i += 256) {
    int t = i / EMB_DIM, k = i - t * EMB_DIM;
    outEmb[(size_t)b * (MAX_LEN * EMB_DIM) + i] =
        embT[(size_t)bagL[t] * EMB_DIM + k];   // row 4096 is the zero pad row
  }
}

// -------------------------------------------------------------------------
// Branch-free tanh: a=|x|, e=exp(-2a), r=(1-e)/(1+e), copysign(r, x).
// Saturates to +/-1 for large |x| (e -> 0), no EXEC divergence, lowers to
// v_exp_f32 + v_rcp_f32 (TRANS pipe, co-executes with WMMA).
// -------------------------------------------------------------------------
static __device__ inline float tanh_fast(float x) {
  const float a = __builtin_fabsf(x);
  const float e = __expf(-2.0f * a);
  const float r = (1.0f - e) * __builtin_amdgcn_rcpf(1.0f + e);
  return __builtin_copysignf(r, x);
}

// -------------------------------------------------------------------------
// Kernel 2: batched RNN via V_WMMA_F32_16X16X4_F32 (exact f32 semantics).
// One wave32 = one tile of 16 samples. P[16x32] = X*Wih^T + H*Whh^T + bias
// is 2 N-tiles x (5 + 8) chained K=4 WMMAs. Weights live in registers in
// B layout; H round-trips a per-wave LDS slab to convert C/D -> A layout.
// -------------------------------------------------------------------------
static __device__ inline v8f wmma4(v2f a, v2f b, v8f c) {
  // (neg_a, A, neg_b, B, c_mod, C, reuse_a, reuse_b)
  return __builtin_amdgcn_wmma_f32_16x16x4_f32(
      false, a, false, b, (short)0, c, false, false);
}

__global__ __launch_bounds__(128) void glyph_rnn_kernel(
    const float* __restrict__ emb,    // [B][64][20]
    const float* __restrict__ Wih,    // [32][20]
    const float* __restrict__ Whh,    // [32][32]
    const float* __restrict__ bih,    // [32]
    const float* __restrict__ bhh,    // [32]
    const int*   __restrict__ lenv,   // [B]
    float* __restrict__ outH)         // [B][32]
{
  __shared__ __align__(16) float Hlds[4][16][HIDDEN];   // per-wave tile of H

  const int lane = threadIdx.x & 31;
  const int wav  = threadIdx.x >> 5;
  const int tile = blockIdx.x * 4 + wav;
  const int tb   = tile * 16;          // first sample in tile
  const int n0   = lane & 15;          // N column within tile / sample row M
  const int kh   = lane >> 4;          // K-half for A/B layouts

  // ---- preload weights in WMMA B layout (lane holds B[2*kh+v][n]) ----
  v2f wih[2][5], whh[2][8];
  float bias[2];
#pragma unroll
  for (int nt = 0; nt < 2; ++nt) {
    const int n = nt * 16 + n0;
    bias[nt] = bih[n] + bhh[n];
#pragma unroll
    for (int kc = 0; kc < 5; ++kc)
      wih[nt][kc] = *reinterpret_cast<const v2f*>(Wih + n * EMB_DIM + kc * 4 + 2 * kh);
#pragma unroll
    for (int kc = 0; kc < 8; ++kc)
      whh[nt][kc] = *reinterpret_cast<const v2f*>(Whh + n * HIDDEN + kc * 4 + 2 * kh);
  }

  int lens[8];
#pragma unroll
  for (int r = 0; r < 8; ++r) lens[r] = lenv[tb + r + 8 * kh];

  // H kept in C/D layout registers AND mirrored in LDS for A-chunk loads.
  float hreg[2][8];
#pragma unroll
  for (int nt = 0; nt < 2; ++nt)
#pragma unroll
    for (int r = 0; r < 8; ++r) {
      hreg[nt][r] = 0.0f;
      Hlds[wav][r + 8 * kh][nt * 16 + n0] = 0.0f;
    }
  __syncthreads();

  const float* xbase = emb + ((size_t)(tb + n0) * MAX_LEN) * EMB_DIM;

  for (int t = 0; t < MAX_LEN; ++t) {
    // A chunks for X_t: lane holds X[m=n0][kc*4 + 2*kh + {0,1}]
    v2f xa[5];
#pragma unroll
    for (int kc = 0; kc < 5; ++kc)
      xa[kc] = *reinterpret_cast<const v2f*>(xbase + (size_t)t * EMB_DIM + kc * 4 + 2 * kh);
    // A chunks for H from LDS
    v2f ha[8];
#pragma unroll
    for (int kc = 0; kc < 8; ++kc)
      ha[kc] = *reinterpret_cast<const v2f*>(&Hlds[wav][n0][kc * 4 + 2 * kh]);

    v8f c0, c1;
#pragma unroll
    for (int r = 0; r < 8; ++r) { c0[r] = bias[0]; c1[r] = bias[1]; }

#pragma unroll
    for (int kc = 0; kc < 5; ++kc) {
      c0 = wmma4(xa[kc], wih[0][kc], c0);
      c1 = wmma4(xa[kc], wih[1][kc], c1);
    }
#pragma unroll
    for (int kc = 0; kc < 8; ++kc) {
      c0 = wmma4(ha[kc], whh[0][kc], c0);
      c1 = wmma4(ha[kc], whh[1][kc], c1);
    }

    __syncthreads();   // all ha loads done before Hlds is overwritten
#pragma unroll
    for (int nt = 0; nt < 2; ++nt)
#pragma unroll
      for (int r = 0; r < 8; ++r) {
        const float acc = (nt == 0) ? c0[r] : c1[r];
        const float hn  = tanh_fast(acc);
        const bool  act = (t < lens[r]);         // freeze past sample length
        hreg[nt][r] = act ? hn : hreg[nt][r];
        Hlds[wav][r + 8 * kh][nt * 16 + n0] = hreg[nt][r];
      }
    __syncthreads();
  }

  // write h: C/D layout element (nt, r) = H[m = r + 8*kh][n = nt*16 + n0]
#pragma unroll
  for (int nt = 0; nt < 2; ++nt)
#pragma unroll
    for (int r = 0; r < 8; ++r)
      outH[(size_t)(tb + r + 8 * kh) * HIDDEN + nt * 16 + n0] = hreg[nt][r];
}

// -------------------------------------------------------------------------
extern "C" void kernel_launch(void* const* d_in, const int* in_sizes, int n_in,
                              void* d_out, int out_size, void* d_ws, size_t ws_size,
                              hipStream_t stream) {
  const int*   chars  = (const int*)  d_in[0];
  const int*   colors = (const int*)  d_in[1];
  const float* embT   = (const float*)d_in[2];
  const float* Wih    = (const float*)d_in[3];
  const float* Whh    = (const float*)d_in[4];
  const float* bih    = (const float*)d_in[5];
  const float* bhh    = (const float*)d_in[6];

  const int B = in_sizes[0] / HW_CELLS;   // 8192

  float* out    = (float*)d_out;
  float* outH   = out;                                        // [B][32]
  float* outEmb = out + (size_t)B * HIDDEN;                   // [B][64][20]
  float* outBag = outEmb + (size_t)B * MAX_LEN * EMB_DIM;     // [B][64]
  int*   lenv   = (int*)d_ws;                                 // [B]

  glyph_bag_kernel<<<B, 256, 0, stream>>>(chars, colors, embT,
                                          outEmb, outBag, lenv);

  const int tiles = B / 16;               // 512 wave-tiles, 4 per block
  glyph_rnn_kernel<<<tiles / 4, 128, 0, stream>>>(outEmb, Wih, Whh,
                                                  bih, bhh, lenv, outH);
}